// BaseGNN_65395172049088
// MI455X (gfx1250) — compile-verified
//
#include <hip/hip_runtime.h>

// GNN: 3x GraphConv (x' = x@Wr + segsum(x[src],dst)@Wn + b) + 64->1 head, clip.
// N=100000, E=1200000, D=H=64.  agg/gather is L2-bound; GEMMs use fp32 WMMA.
#define N_NODES 100000
#define E_EDGES 1200000
#define HDIM    64
#define MTILES  (N_NODES / 16)   // 6250, exact
#define KSTEPS  (HDIM / 4)       // 16 WMMA K-steps of 4
#define NTILES  (HDIM / 16)      // 4 output column tiles

typedef __attribute__((ext_vector_type(2))) float v2f;
typedef __attribute__((ext_vector_type(8))) float v8f;

// ---------------------------------------------------------------- zero agg
__global__ void gnn_zero_kernel(float4* __restrict__ p, int n4) {
    int i = blockIdx.x * blockDim.x + threadIdx.x;
    if (i < n4) p[i] = make_float4(0.f, 0.f, 0.f, 0.f);
}

// ------------------------------------------------- edge scatter-aggregate
// 16 threads per edge; each thread moves one float4 (4 features) and does
// 4 fp32 global atomic adds (no-return -> global_atomic_add_f32, STOREcnt).
__global__ __launch_bounds__(256) void gnn_scatter_kernel(
    const float* __restrict__ x,
    const int*   __restrict__ src,
    const int*   __restrict__ dst,
    float*       __restrict__ agg) {
    int gid = blockIdx.x * blockDim.x + threadIdx.x;
    int e = gid >> 4;
    if (e >= E_EDGES) return;
    int f = (gid & 15) << 2;
    int s = src[e];
    int d = dst[e];
    const float4 v = *(const float4*)(x + (long)s * HDIM + f);
    float* o = agg + (long)d * HDIM + f;
    __hip_atomic_fetch_add(o + 0, v.x, __ATOMIC_RELAXED, __HIP_MEMORY_SCOPE_AGENT);
    __hip_atomic_fetch_add(o + 1, v.y, __ATOMIC_RELAXED, __HIP_MEMORY_SCOPE_AGENT);
    __hip_atomic_fetch_add(o + 2, v.z, __ATOMIC_RELAXED, __HIP_MEMORY_SCOPE_AGENT);
    __hip_atomic_fetch_add(o + 3, v.w, __ATOMIC_RELAXED, __HIP_MEMORY_SCOPE_AGENT);
}

// ------------------------------------------------------- fused dual GEMM
// Y[16m..16m+15, 0..63] = X_tile @ Wr + G_tile @ Wn + b
// One wave per 16-row strip; KSTEPS x NTILES x 2 sources
// = 128 v_wmma_f32_16x16x4_f32 per wave.
// Weights are staged in LDS pre-swizzled into per-lane B-fragment order so
// the inner loop is a single aligned ds_load_b64 per fragment (no repacking
// movs); consecutive lanes hit consecutive 8B slots -> bank-conflict free.
__global__ __launch_bounds__(256) void gnn_gemm_kernel(
    const float* __restrict__ X,   // [N,64] node features
    const float* __restrict__ G,   // [N,64] aggregated neighbor features
    const float* __restrict__ Wr,  // [64,64] row-major (in,out)
    const float* __restrict__ Wn,  // [64,64]
    const float* __restrict__ bias,// [64]
    float*       __restrict__ Y) { // [N,64]
    // sW[(step*NTILES + n)*32 + lane] = { W[(4*step+2*half)*64 + n*16+lrow],
    //                                    W[(4*step+2*half+1)*64 + n*16+lrow] }
    __shared__ v2f  sWr[KSTEPS * NTILES * 32];   // 16 KB
    __shared__ v2f  sWn[KSTEPS * NTILES * 32];   // 16 KB
    __shared__ float sb[HDIM];

    const int tid = threadIdx.x;
    for (int idx = tid; idx < KSTEPS * NTILES * 32; idx += 256) {
        const int step = idx >> 7;          // 0..15
        const int rem  = idx & 127;
        const int n    = rem >> 5;          // 0..3
        const int ln   = rem & 31;
        const int hf   = ln >> 4;
        const int lr   = ln & 15;
        const int g = (step * 4 + 2 * hf) * HDIM + n * 16 + lr;
        sWr[idx] = v2f{ Wr[g], Wr[g + HDIM] };
        sWn[idx] = v2f{ Wn[g], Wn[g + HDIM] };
    }
    if (tid < HDIM) sb[tid] = bias[tid];
    __syncthreads();

    const int wave = tid >> 5;
    const int m = blockIdx.x * 8 + wave;          // row-tile index (wave-uniform)
    if (m >= MTILES) return;                      // whole wave exits uniformly

    const int lane = tid & 31;
    const int half = lane >> 4;                   // 0: lanes 0-15, 1: lanes 16-31
    const int lrow = lane & 15;

    // A fragment (16x4 f32): lane = M, VGPR = K-lo, lane-half = K-pair -> float2
    const long arow = (long)(m * 16 + lrow) * HDIM + 2 * half;

    v8f acc[NTILES] = {};                         // 4 N-tiles of 16x16 f32

#pragma unroll 4
    for (int step = 0; step < KSTEPS; ++step) {
        const v2f ax = *(const v2f*)(X + arow + step * 4);
        const v2f ag = *(const v2f*)(G + arow + step * 4);
        const v2f* brp = &sWr[(step * NTILES) * 32 + lane];
        const v2f* bnp = &sWn[(step * NTILES) * 32 + lane];
#pragma unroll
        for (int n = 0; n < NTILES; ++n) {
            const v2f br = brp[n * 32];           // aligned ds_load_b64
            const v2f bn = bnp[n * 32];
            acc[n] = __builtin_amdgcn_wmma_f32_16x16x4_f32(
                false, ax, false, br, (short)0, acc[n], false, false);
            acc[n] = __builtin_amdgcn_wmma_f32_16x16x4_f32(
                false, ag, false, bn, (short)0, acc[n], false, false);
        }
    }

    // C/D layout: VGPR j -> row M = j + 8*half, col N = lane%16 (+16n)
#pragma unroll
    for (int n = 0; n < NTILES; ++n) {
        const int col = n * 16 + lrow;
        const float bv = sb[col];
#pragma unroll
        for (int j = 0; j < 8; ++j) {
            const int row = m * 16 + j + 8 * half;
            Y[(long)row * HDIM + col] = acc[n][j] + bv;
        }
    }
}

// ------------------------------------------------------------ output head
__global__ void gnn_head_kernel(const float* __restrict__ X,
                                const float* __restrict__ Wout,
                                const float* __restrict__ bout,
                                float* __restrict__ out) {
    int i = blockIdx.x * blockDim.x + threadIdx.x;
    if (i >= N_NODES) return;
    const float4* xr = (const float4*)(X + (long)i * HDIM);
    const float4* wr = (const float4*)Wout;
    float s = bout[0];
#pragma unroll
    for (int f = 0; f < 16; ++f) {
        const float4 a = xr[f];
        const float4 w = wr[f];
        s += a.x * w.x + a.y * w.y + a.z * w.z + a.w * w.w;
    }
    out[i] = fminf(4.0f, fmaxf(-4.0f, s));
}

// ---------------------------------------------------------------- launcher
extern "C" void kernel_launch(void* const* d_in, const int* in_sizes, int n_in,
                              void* d_out, int out_size, void* d_ws, size_t ws_size,
                              hipStream_t stream) {
    // setup_inputs() order:
    // 0:x 1:edge_index 2:Wr0 3:Wn0 4:b0 5:Wr1 6:Wn1 7:b1 8:Wr2 9:Wn2 10:b2 11:Wout 12:bout
    const float* x    = (const float*)d_in[0];
    const int*   ei   = (const int*)d_in[1];          // harness: integer -> const int*
    const int*   src  = ei;
    const int*   dst  = ei + E_EDGES;
    const float* Wr[3] = {(const float*)d_in[2], (const float*)d_in[5], (const float*)d_in[8]};
    const float* Wn[3] = {(const float*)d_in[3], (const float*)d_in[6], (const float*)d_in[9]};
    const float* bs[3] = {(const float*)d_in[4], (const float*)d_in[7], (const float*)d_in[10]};
    const float* Wout = (const float*)d_in[11];
    const float* bout = (const float*)d_in[12];
    float* out = (float*)d_out;

    // workspace: agg | bufA | bufB  (3 * 25.6 MB)
    const size_t feat = (size_t)N_NODES * HDIM;
    float* agg  = (float*)d_ws;
    float* bufA = agg + feat;
    float* bufB = bufA + feat;

    const dim3 b256(256);
    const int zeroBlocks    = (int)(feat / 4 / 256);        // 6250 exact
    const int scatterBlocks = (E_EDGES * 16) / 256;         // 75000 exact
    const int gemmBlocks    = (MTILES + 7) / 8;             // 782

    const float* cur = x;
    float* nxt[3] = {bufA, bufB, bufA};
    for (int l = 0; l < 3; ++l) {
        gnn_zero_kernel<<<zeroBlocks, b256, 0, stream>>>((float4*)agg, (int)(feat / 4));
        gnn_scatter_kernel<<<scatterBlocks, b256, 0, stream>>>(cur, src, dst, agg);
        gnn_gemm_kernel<<<gemmBlocks, b256, 0, stream>>>(cur, agg, Wr[l], Wn[l], bs[l], nxt[l]);
        cur = nxt[l];
    }
    gnn_head_kernel<<<(N_NODES + 255) / 256, b256, 0, stream>>>(cur, Wout, bout, out);
}